// EMAFusion_49039936585905
// MI455X (gfx1250) — compile-verified
//
#include <hip/hip_runtime.h>

typedef __attribute__((ext_vector_type(2))) float v2f;
typedef __attribute__((ext_vector_type(8))) float v8f;

// EMAFusion: h_t = (1-a)*y_t + a*h_{t-1}, a = 0.9, y_seq (B,S,D) fp32.
// Chunked-scan-as-matmul: 16-step chunks -> D = L*Y + diag(a^{t+1})*carry
// with L lower-triangular, executed as 4x chained V_WMMA_F32_16X16X4_F32.
// Bandwidth-bound: 268 MB total traffic, ~11.5 us floor at 23.3 TB/s; input
// (134 MB) fits in 192 MB L2. Row-spread WGP-scope prefetch (inline asm,
// scope 0 = pull into ALL cache levels) keeps the 8-load burst at near-L0
// latency in steady state.

constexpr int kB = 4;
constexpr int kS = 4096;
constexpr int kD = 2048;
constexpr int kTS = 16;                   // WMMA tile: 16 time steps x 16 channels
constexpr int kStripsPerB = kD / 16;      // 128
constexpr int kStrips = kB * kStripsPerB; // 512 waves total
constexpr int kChunks = kS / kTS;         // 256 sequential chunks per strip

#define ALPHA_F 0.9f

// WGP-scope (scope 0) prefetch: scope 0 fills every cache level on a miss.
// __builtin_prefetch only emits SE/SYS scope,
// which skips the WGP cache, so use inline asm with the default scope.
__device__ __forceinline__ void prefetch_wgp(const void* p) {
  asm volatile("global_prefetch_b8 %0, off" :: "v"(p) : "memory");
}

__global__ __launch_bounds__(256, 1) void ema_scan_wmma(
    const float* __restrict__ y, float* __restrict__ out) {
  const int gtid = blockIdx.x * blockDim.x + threadIdx.x;
  const int wave = gtid >> 5;
  const int lane = threadIdx.x & 31;

  const int bidx   = wave / kStripsPerB;
  const int col0   = (wave % kStripsPerB) * 16;
  const int n      = lane & 15;            // channel (N) / row (M) within tile
  const int khalf  = (lane < 16) ? 0 : 2;  // A/B layout: hi half-lanes hold K+2
  const int rowoff = (lane < 16) ? 0 : 8;  // C/D layout: hi half-lanes hold M+8

  const float one_m_a = 1.0f - ALPHA_F;
  const float l2a = -0.15200309344504997f; // log2(0.9)

  // A-matrix = L: L[t][k] = (t>=k) ? (1-a)*a^(t-k) : 0, as four 16x4 K-groups.
  // 16x4 f32 A layout: lane m holds row M=m; VGPR0 -> K = k0, VGPR1 -> K = k0+1,
  // where k0 = 4j + (lane<16 ? 0 : 2).
  v2f a[4];
  #pragma unroll
  for (int j = 0; j < 4; ++j) {
    const int k0 = 4 * j + khalf;
    const int k1 = k0 + 1;
    a[j].x = (n >= k0) ? one_m_a * exp2f((float)(n - k0) * l2a) : 0.0f;
    a[j].y = (n >= k1) ? one_m_a * exp2f((float)(n - k1) * l2a) : 0.0f;
  }

  // Carry-injection coefficients folded into the accumulator:
  // C[r] = a^(t+1) * carry, t = rowoff + r.
  float coef[8];
  #pragma unroll
  for (int r = 0; r < 8; ++r)
    coef[r] = exp2f((float)(rowoff + r + 1) * l2a);

  const size_t strip_base = (size_t)bidx * kS * kD + col0;
  const float* yb = y + strip_base + n;
  float*       ob = out + strip_base + n;
  // Prefetch stream: lane l covers row tbase+l -> one global_prefetch_b8
  // touches 32 distinct cachelines = the full footprint of TWO future chunks
  // (rows are 8 KB apart, each in its own line).
  const float* ypf = y + strip_base + (size_t)lane * kD;

  float carry = 0.0f;

  // Register double-buffer for the B-tile (16 time steps x 16 channels).
  // B 4x16 layout mirrors A: VGPR0 -> K = 4j+khalf, VGPR1 -> K = 4j+khalf+1.
  v2f bt[4], bn[4];
  #pragma unroll
  for (int j = 0; j < 4; ++j) {
    const size_t r0 = (size_t)(4 * j + khalf) * kD;
    bt[j].x = yb[r0];
    bt[j].y = yb[r0 + kD];
  }
  // Warm the first few chunks into WGP$/L2.
  prefetch_wgp(ypf);                          // chunks 0-1
  prefetch_wgp(ypf + (size_t)2 * kTS * kD);   // chunks 2-3
  prefetch_wgp(ypf + (size_t)4 * kTS * kD);   // chunks 4-5

  for (int ci = 0; ci < kChunks; ++ci) {
    // Stage next tile (register double buffer) while this one computes.
    if (ci + 1 < kChunks) {
      const float* yn = yb + (size_t)(ci + 1) * kTS * kD;
      #pragma unroll
      for (int j = 0; j < 4; ++j) {
        const size_t r0 = (size_t)(4 * j + khalf) * kD;
        bn[j].x = yn[r0];
        bn[j].y = yn[r0 + kD];
      }
    }
    // Every other iteration, prefetch the next pair of chunks 6 ahead
    // (lane-spread across rows; WGP scope -> all cache levels).
    if (((ci & 1) == 0) && (ci + 7 < kChunks))
      prefetch_wgp(ypf + (size_t)(ci + 6) * kTS * kD);

    // Accumulator pre-loaded with the decayed carry-in.
    v8f c;
    #pragma unroll
    for (int r = 0; r < 8; ++r) c[r] = coef[r] * carry;

    // D = L * Y_chunk + C, chained over K = 16 in four K=4 steps.
    c = __builtin_amdgcn_wmma_f32_16x16x4_f32(false, a[0], false, bt[0], (short)0, c, false, false);
    c = __builtin_amdgcn_wmma_f32_16x16x4_f32(false, a[1], false, bt[1], (short)0, c, false, false);
    c = __builtin_amdgcn_wmma_f32_16x16x4_f32(false, a[2], false, bt[2], (short)0, c, false, false);
    c = __builtin_amdgcn_wmma_f32_16x16x4_f32(false, a[3], false, bt[3], (short)0, c, false, false);

    // Store the 16x16 result tile (C/D layout: VGPR r -> row rowoff+r).
    float* op = ob + (size_t)ci * kTS * kD;
    #pragma unroll
    for (int r = 0; r < 8; ++r)
      op[(size_t)(rowoff + r) * kD] = c[r];

    // New carry = h at t=15 for channel n: lives in c[7] of lane 16+n.
    // Broadcast to all lanes with a wave32 backward permute.
    carry = __int_as_float(
        __builtin_amdgcn_ds_bpermute((16 + n) << 2, __float_as_int(c[7])));

    #pragma unroll
    for (int j = 0; j < 4; ++j) bt[j] = bn[j];
  }
}

extern "C" void kernel_launch(void* const* d_in, const int* in_sizes, int n_in,
                              void* d_out, int out_size, void* d_ws, size_t ws_size,
                              hipStream_t stream) {
  (void)in_sizes; (void)n_in; (void)out_size; (void)d_ws; (void)ws_size;
  const float* y = (const float*)d_in[0];
  float* out = (float*)d_out;
  // 512 waves, one per 16-channel strip: 64 blocks x 256 threads (8 waves/block).
  const int threads = 256;
  const int blocks = (kStrips * 32) / threads; // 64
  ema_scan_wmma<<<blocks, threads, 0, stream>>>(y, out);
}